// KuramotoLayer_15925738734062
// MI455X (gfx1250) — compile-verified
//
#include <hip/hip_runtime.h>
#include <math.h>

typedef float v2f __attribute__((ext_vector_type(2)));
typedef float v8f __attribute__((ext_vector_type(8)));

#define NOSC 2048
#define BATCH 32
#define DT 0.1f
#define STEPS 10

// S/C storage layout: pair-interleaved so a WMMA B-fragment is ONE b64 load.
//   element (j, b)  ->  S[(j>>1)*64 + b*2 + (j&1)]
// For an even k-row kk, the fragment pair {S[kk,b], S[kk+1,b]} is contiguous
// and 8-byte aligned.
__device__ __forceinline__ size_t sc_idx(int j, int b) {
    return (size_t)(j >> 1) * (2 * BATCH) + (size_t)b * 2 + (j & 1);
}

// ---------------------------------------------------------------------------
// One-time prep: copy theta into workspace (don't mutate inputs), and build
// pair-interleaved sin/cos arrays.
// ---------------------------------------------------------------------------
__global__ __launch_bounds__(256)
void kur_prep(const float* __restrict__ theta_in,
              float* __restrict__ th,
              float* __restrict__ S,
              float* __restrict__ C)
{
    int idx = blockIdx.x * 256 + threadIdx.x;   // 0 .. 65535
    int b = idx >> 11;            // batch
    int i = idx & (NOSC - 1);     // oscillator
    float t = theta_in[idx];
    th[idx] = t;
    float s = sinf(t), c = cosf(t);
    S[sc_idx(i, b)] = s;
    C[sc_idx(i, b)] = c;
}

// ---------------------------------------------------------------------------
// One Kuramoto step, fused:  P1 = K @ S, P2 = K @ C  via fp32 WMMA, then the
// Euler update + wrap + next-step sin/cos, all in the tile-owning wave.
//
// Grid: 16 blocks x 256 threads = 128 waves. Wave w owns M-tile mtile
// (rows [16*mtile, 16*mtile+16) of K) and accumulates 4 16x16 f32 tiles:
//   {sin,cos} x {batch 0..15, batch 16..31}
// K-loop: k += 4 (V_WMMA_F32_16X16X4_F32), 512 iterations, unrolled 8x.
// Per iteration: 5x global_load_b64 feeding 4 WMMAs.
//
// Fragment layouts (CDNA5 ISA 7.12.2, wave32):
//   A 16x4 f32 : lane L: row = base + L%16 ; VGPR0/1 = k + 2*(L/16), +1
//   B 4x16 f32 : lane L: col = L%16 ; VGPR0/1 = k-rows (k + 2*(L/16)), +1
//   D 16x16 f32: VGPR v, lane L: i = base + v + 8*(L/16), col = L%16
// ---------------------------------------------------------------------------
__global__ __launch_bounds__(256)
void kur_step(const float* __restrict__ Kmat,
              const float* __restrict__ omega,
              const float* __restrict__ Kglob,
              const float* __restrict__ mug,
              float* __restrict__ th,
              const float* __restrict__ Sin,
              const float* __restrict__ Cin,
              float* __restrict__ Sout,
              float* __restrict__ Cout)
{
    const int lane  = threadIdx.x & 31;
    const int wave  = threadIdx.x >> 5;
    const int mtile = blockIdx.x * 8 + wave;   // 0..127
    const int half  = lane >> 4;               // 0 or 1
    const int l16   = lane & 15;

    // A-fragment base: row (mtile*16 + l16), k-offset 2*half (b64-aligned).
    const float* Arow = Kmat + (size_t)(mtile * 16 + l16) * NOSC + 2 * half;
    // B-fragment base for k-row (k + 2*half), column l16 (ntile 0):
    //   sc_idx(k + 2*half, l16) = ((k>>1) + half)*64 + l16*2
    const float* Bs = Sin + (size_t)half * (2 * BATCH) + (size_t)l16 * 2;
    const float* Bc = Cin + (size_t)half * (2 * BATCH) + (size_t)l16 * 2;

    v8f ps0 = {}, ps1 = {}, pc0 = {}, pc1 = {};

    #pragma unroll 8
    for (int k = 0; k < NOSC; k += 4) {
        size_t bo = (size_t)(k >> 1) * (2 * BATCH);
        v2f a   = *(const v2f*)(Arow + k);          // K[row, kk .. kk+1]
        v2f bs0 = *(const v2f*)(Bs + bo);           // S[kk..kk+1, col]
        v2f bs1 = *(const v2f*)(Bs + bo + 32);      // S[kk..kk+1, col+16]
        v2f bc0 = *(const v2f*)(Bc + bo);
        v2f bc1 = *(const v2f*)(Bc + bo + 32);
        ps0 = __builtin_amdgcn_wmma_f32_16x16x4_f32(false, a, false, bs0, (short)0, ps0, false, false);
        ps1 = __builtin_amdgcn_wmma_f32_16x16x4_f32(false, a, false, bs1, (short)0, ps1, false, false);
        pc0 = __builtin_amdgcn_wmma_f32_16x16x4_f32(false, a, false, bc0, (short)0, pc0, false, false);
        pc1 = __builtin_amdgcn_wmma_f32_16x16x4_f32(false, a, false, bc1, (short)0, pc1, false, false);
    }

    // Fused update: th += DT*(omega + coef * mus * (cos_i*P1 - sin_i*P2)), wrap.
    const float coef = Kglob[0] * (1.0f / (float)NOSC);  // K_global / N
    const float mus  = mug[0] * 0.5f;                    // mu_gate * MU_BRANCH_SCALE

    #pragma unroll
    for (int v = 0; v < 8; ++v) {
        int i = mtile * 16 + v + 8 * half;
        float om = omega[i];
        float p1[2] = { ps0[v], ps1[v] };
        float p2[2] = { pc0[v], pc1[v] };
        #pragma unroll
        for (int nt = 0; nt < 2; ++nt) {
            int b = nt * 16 + l16;
            float s_i = Sin[sc_idx(i, b)];
            float c_i = Cin[sc_idx(i, b)];
            float coupling = c_i * p1[nt] - s_i * p2[nt];
            float dth  = om + coef * (mus * coupling);
            float tnew = th[(size_t)b * NOSC + i] + DT * dth;
            float sn = sinf(tnew), cn = cosf(tnew);
            th[(size_t)b * NOSC + i] = atan2f(sn, cn);   // wrap to (-pi, pi]
            Sout[sc_idx(i, b)] = sn;                     // sin(wrap(t)) == sn
            Cout[sc_idx(i, b)] = cn;
        }
    }
}

// ---------------------------------------------------------------------------
// Coherence per batch (final sin/cos already sit in S/C) + theta copy-out.
// Output layout: theta[32*2048] then coherence[32].
// ---------------------------------------------------------------------------
__global__ __launch_bounds__(256)
void kur_finalize(const float* __restrict__ th,
                  const float* __restrict__ S,
                  const float* __restrict__ C,
                  float* __restrict__ out)
{
    __shared__ float ssum[256];
    __shared__ float csum[256];
    int b = blockIdx.x;
    int tid = threadIdx.x;
    float sa = 0.f, ca = 0.f;
    for (int i = tid; i < NOSC; i += 256) {
        sa += S[sc_idx(i, b)];
        ca += C[sc_idx(i, b)];
        out[(size_t)b * NOSC + i] = th[(size_t)b * NOSC + i];
    }
    ssum[tid] = sa; csum[tid] = ca;
    __syncthreads();
    for (int off = 128; off > 0; off >>= 1) {
        if (tid < off) { ssum[tid] += ssum[tid + off]; csum[tid] += csum[tid + off]; }
        __syncthreads();
    }
    if (tid == 0) {
        float sm = ssum[0] * (1.0f / (float)NOSC);
        float cm = csum[0] * (1.0f / (float)NOSC);
        out[(size_t)BATCH * NOSC + b] = sqrtf(sm * sm + cm * cm);
    }
}

// ---------------------------------------------------------------------------
extern "C" void kernel_launch(void* const* d_in, const int* in_sizes, int n_in,
                              void* d_out, int out_size, void* d_ws, size_t ws_size,
                              hipStream_t stream)
{
    const float* theta = (const float*)d_in[0];   // (32, 2048)
    const float* Kmat  = (const float*)d_in[1];   // (2048, 2048)
    const float* omega = (const float*)d_in[2];   // (2048,)
    const float* Kglob = (const float*)d_in[3];   // scalar
    const float* mug   = (const float*)d_in[4];   // (1,)
    float* out = (float*)d_out;

    const size_t BN = (size_t)BATCH * NOSC;       // 65536
    float* th = (float*)d_ws;                     // current theta (b-major)
    float* S0 = th + BN;                          // sin, pair-interleaved
    float* C0 = S0 + BN;
    float* S1 = C0 + BN;                          // double buffer
    float* C1 = S1 + BN;

    kur_prep<<<(int)(BN / 256), 256, 0, stream>>>(theta, th, S0, C0);

    for (int t = 0; t < STEPS; ++t) {
        const float* Si = (t & 1) ? S1 : S0;
        const float* Ci = (t & 1) ? C1 : C0;
        float*       So = (t & 1) ? S0 : S1;
        float*       Co = (t & 1) ? C0 : C1;
        kur_step<<<16, 256, 0, stream>>>(Kmat, omega, Kglob, mug, th, Si, Ci, So, Co);
    }

    // After 10 steps (even), final sin/cos are back in S0/C0.
    kur_finalize<<<BATCH, 256, 0, stream>>>(th, S0, C0, out);
}